// GCCGC_32006096290300
// MI455X (gfx1250) — compile-verified
//
#include <hip/hip_runtime.h>
#include <math.h>

// MI455X / gfx1250, wave32. All heavy math via V_WMMA_F32_16X16X4_F32 (fp32
// matrix pipe — inputs are fp32, problem is bandwidth-bound, so no reason to
// drop precision to bf16).

typedef __attribute__((ext_vector_type(2))) float v2f;
typedef __attribute__((ext_vector_type(8))) float v8f;

#define Bn   32
#define Cn   64
#define Tn   128
#define Vn   25
#define OUTn 64
#define RELn 8
#define TSn  9

// ---------------------------------------------------------------------------
// Kernel 1: x1[b,r,v], x2[b,r,v] = w{1,2} @ mean_t(x) + b{1,2}
// One block per b. xm (C x V = 1600 floats) staged in LDS.
// ---------------------------------------------------------------------------
__global__ __launch_bounds__(256) void prep_kernel(
    const float* __restrict__ x,
    const float* __restrict__ w1, const float* __restrict__ b1,
    const float* __restrict__ w2, const float* __restrict__ b2,
    float* __restrict__ x1ws, float* __restrict__ x2ws)
{
  __shared__ float xm[Cn * Vn];
  const int tid = threadIdx.x;
  const int b   = blockIdx.x;

  for (int e = tid; e < Cn * Vn; e += 256) {
    int c = e / Vn, v = e - c * Vn;
    const float* p = x + ((size_t)(b * Cn + c) * Tn) * Vn + v;
    float s = 0.f;
    for (int t = 0; t < Tn; ++t) s += p[t * Vn];
    xm[e] = s * (1.0f / Tn);
  }
  __syncthreads();

  for (int e = tid; e < 2 * RELn * Vn; e += 256) {
    int mat = e / (RELn * Vn);
    int rem = e - mat * RELn * Vn;
    int r = rem / Vn, v = rem - r * Vn;
    const float* wr = (mat ? w2 : w1) + r * Cn;
    float s = 0.f;
    for (int c = 0; c < Cn; ++c) s += wr[c] * xm[c * Vn + v];
    s += (mat ? b2[r] : b1[r]);
    (mat ? x2ws : x1ws)[b * RELn * Vn + r * Vn + v] = s;
  }
}

// ---------------------------------------------------------------------------
// Kernel 2: z[b,c,t,i] = sum_{k,j} x[b,c,t+k-8,j] * W[b,c,k,j,i]
// where W[b,c,k,j,i] = sum_r w4[c*9+k,r]*tanh(x1[b,r,j]-x2[b,r,i]) + b4 + A[j,i]
//
// One 512-thread block (16 wave32) per (b,c). W tile (22.5 KB) is built in
// LDS and never touches memory (saves a 46 MB round-trip). x slab is staged
// causally padded. Pads are zero-filled so the WMMA loop has no conditionals
// (EXEC must stay all-ones). Each wave owns one 16x16 (t x i) tile:
// 9 k-steps x 7 K-chunks = 63 v_wmma_f32_16x16x4_f32 per wave.
// ---------------------------------------------------------------------------
__global__ __launch_bounds__(512) void zdyn_kernel(
    const float* __restrict__ x, const float* __restrict__ Aadj,
    const float* __restrict__ w4, const float* __restrict__ b4,
    const float* __restrict__ x1ws, const float* __restrict__ x2ws,
    float* __restrict__ zws)
{
  __shared__ float xpad[136 * 28];        // rows -8..127 of x (zero-padded), cols padded to 28
  __shared__ float Wlds[TSn * 28 * 32];   // [k][j(pad28)][i(pad32)]
  __shared__ float x1s[RELn * Vn];
  __shared__ float x2s[RELn * Vn];

  const int tid = threadIdx.x;
  const int bc  = blockIdx.x;
  const int b   = bc >> 6;
  const int c   = bc & 63;

  // --- fill phase -----------------------------------------------------
  for (int e = tid; e < TSn * 28 * 32; e += 512) Wlds[e] = 0.f;

  const float* xb = x + ((size_t)(b * Cn + c) * Tn) * Vn;
  for (int e = tid; e < 136 * 28; e += 512) {
    int row = e / 28, col = e - row * 28;
    float v = 0.f;
    if (col < Vn && row >= (TSn - 1)) v = xb[(row - (TSn - 1)) * Vn + col];
    xpad[e] = v;
  }
  if (tid < RELn * Vn) {
    x1s[tid] = x1ws[b * RELn * Vn + tid];
    x2s[tid] = x2ws[b * RELn * Vn + tid];
  }
  __syncthreads();

  // --- W tile: tanh once per (j,i), reused across all 9 k ---------------
  for (int p = tid; p < Vn * Vn; p += 512) {
    int j = p / Vn, i = p - j * Vn;
    float th[RELn];
#pragma unroll
    for (int r = 0; r < RELn; ++r)
      th[r] = tanhf(x1s[r * Vn + j] - x2s[r * Vn + i]);
    float aji = Aadj[j * Vn + i];
#pragma unroll
    for (int k = 0; k < TSn; ++k) {
      int o = c * TSn + k;
      float acc = b4[o] + aji;
#pragma unroll
      for (int r = 0; r < RELn; ++r) acc += w4[o * RELn + r] * th[r];
      Wlds[(k * 28 + j) * 32 + i] = acc;
    }
  }
  __syncthreads();

  // --- WMMA phase -------------------------------------------------------
  const int wave  = tid >> 5;
  const int lane  = tid & 31;
  const int laneM = lane & 15;       // M (A) / N (B,C,D) index within tile
  const int hi    = lane >> 4;       // half-wave select for K striping
  const int mt    = wave >> 1;       // t-tile 0..7
  const int nt    = wave & 1;        // i-tile 0..1
  const int t0    = mt * 16;
  const int i0    = nt * 16;

  v8f acc = {};
  for (int k = 0; k < TSn; ++k) {
    const float* arow = &xpad[(t0 + laneM + k) * 28];
#pragma unroll
    for (int jc = 0; jc < 7; ++jc) {
      int j = jc * 4 + 2 * hi;                     // K-pair base for this half-wave
      v2f a = *reinterpret_cast<const v2f*>(&arow[j]);   // ds_load_b64 (8B aligned)
      v2f bb;
      bb.x = Wlds[(k * 28 + j)     * 32 + i0 + laneM];
      bb.y = Wlds[(k * 28 + j + 1) * 32 + i0 + laneM];
      acc = __builtin_amdgcn_wmma_f32_16x16x4_f32(
          false, a, false, bb, (short)0, acc, false, false);
    }
  }

  // --- store z (C/D layout: M = r + 8*hi, N = laneM) --------------------
  float* zb = zws + ((size_t)(b * Cn + c) * Tn) * Vn;
  int i = i0 + laneM;
  if (i < Vn) {
#pragma unroll
    for (int r = 0; r < 8; ++r) {
      int t = t0 + r + 8 * hi;
      zb[t * Vn + i] = acc[r];
    }
  }
}

// ---------------------------------------------------------------------------
// Kernel 3: out[b,o,t,v] = sum_c w3[o,c] * z[b,c,t,v] + b3[o]
// Per-b GEMM: M=64 (o), K=64 (c), N=3200 (t*V contiguous -> coalesced).
// One wave per 16x16 output tile; K = 16 chunks of 4 -> 16 f32 WMMAs.
// ---------------------------------------------------------------------------
__global__ __launch_bounds__(256) void outc_kernel(
    const float* __restrict__ zws, const float* __restrict__ w3,
    const float* __restrict__ b3, float* __restrict__ out)
{
  const int tid   = threadIdx.x;
  const int wave  = tid >> 5;
  const int lane  = tid & 31;
  const int laneM = lane & 15;
  const int hi    = lane >> 4;

  const int task = blockIdx.x * 8 + wave;          // 32 b * 4 ot * 200 nt = 25600
  const int b    = task / 800;
  const int rem  = task - b * 800;
  const int ot   = rem / 200;
  const int nt   = rem - ot * 200;
  const int o0   = ot * 16;
  const int n0   = nt * 16;

  const float* zb = zws + (size_t)b * Cn * Tn * Vn;   // [c][3200]

  v8f acc = {};
#pragma unroll
  for (int kk = 0; kk < 16; ++kk) {
    int cb = kk * 4 + 2 * hi;
    v2f a;                                          // A = w3[o, c] (pair contiguous)
    a.x = w3[(o0 + laneM) * Cn + cb];
    a.y = w3[(o0 + laneM) * Cn + cb + 1];
    v2f bb;                                         // B = z[c, n] (lane-contiguous N)
    bb.x = zb[(size_t)cb       * (Tn * Vn) + n0 + laneM];
    bb.y = zb[(size_t)(cb + 1) * (Tn * Vn) + n0 + laneM];
    acc = __builtin_amdgcn_wmma_f32_16x16x4_f32(
        false, a, false, bb, (short)0, acc, false, false);
  }

#pragma unroll
  for (int r = 0; r < 8; ++r) {
    int o = o0 + r + 8 * hi;
    out[((size_t)(b * OUTn + o)) * (Tn * Vn) + n0 + laneM] = acc[r] + b3[o];
  }
}

// ---------------------------------------------------------------------------
extern "C" void kernel_launch(void* const* d_in, const int* in_sizes, int n_in,
                              void* d_out, int out_size, void* d_ws, size_t ws_size,
                              hipStream_t stream) {
  const float* x  = (const float*)d_in[0];
  const float* A  = (const float*)d_in[1];
  const float* w1 = (const float*)d_in[2];
  const float* b1 = (const float*)d_in[3];
  const float* w2 = (const float*)d_in[4];
  const float* b2 = (const float*)d_in[5];
  const float* w3 = (const float*)d_in[6];
  const float* b3 = (const float*)d_in[7];
  const float* w4 = (const float*)d_in[8];
  const float* b4 = (const float*)d_in[9];
  float* out = (float*)d_out;

  // workspace: x1 | x2 | z   (~26.3 MB total)
  float* x1ws = (float*)d_ws;
  float* x2ws = x1ws + Bn * RELn * Vn;
  float* zws  = x2ws + Bn * RELn * Vn;

  prep_kernel<<<Bn, 256, 0, stream>>>(x, w1, b1, w2, b2, x1ws, x2ws);
  zdyn_kernel<<<Bn * Cn, 512, 0, stream>>>(x, A, w4, b4, x1ws, x2ws, zws);
  outc_kernel<<<(Bn * 4 * 200) / 8, 256, 0, stream>>>(zws, w3, b3, out);
}